// out_heads_with_generic_72988674228879
// MI455X (gfx1250) — compile-verified
//
#include <hip/hip_runtime.h>
#include <stdint.h>

// Problem constants (match reference)
constexpr int Bn  = 16384;
constexpr int IND = 1024;
constexpr int HID = 2048;
constexpr int OUTD = 256;
constexpr int Gn  = 8;
constexpr int TM  = 16;              // rows per M-tile / workgroup
constexpr int LXS = IND + 4;         // padded LDS row stride for X (bank-conflict avoidance)
constexpr int LHS = HID + 4;         // padded LDS row stride for H

typedef __attribute__((ext_vector_type(2))) float v2f;
typedef __attribute__((ext_vector_type(8))) float v8f;

__device__ __forceinline__ v8f wmma_f32(v2f a, v2f b, v8f c) {
  // D = A(16x4, f32) x B(4x16, f32) + C(16x16, f32)
  return __builtin_amdgcn_wmma_f32_16x16x4_f32(
      /*neg_a=*/false, a, /*neg_b=*/false, b,
      /*c_mod=*/(short)0, c, /*reuse_a=*/false, /*reuse_b=*/false);
}

// Async direct-to-LDS copy of one 16B chunk per lane (CDNA5 GLOBAL_LOAD_ASYNC_TO_LDS_B128,
// tracked by ASYNCcnt). ldsOff = wave-relative LDS byte offset (low 32 bits of a generic
// pointer to __shared__), gsrc = per-lane 64-bit global address.
__device__ __forceinline__ void async_load_lds_b128(unsigned ldsOff, const void* gsrc) {
  asm volatile("global_load_async_to_lds_b128 %0, %1, off"
               :: "v"(ldsOff), "v"(gsrc)
               : "memory");
}

__device__ __forceinline__ void wait_asynccnt0() {
  asm volatile("s_wait_asynccnt 0x0" ::: "memory");
}

// ---------------------------------------------------------------------------
// Kernel 1: counting-sort rows by group id.
//   perm[ rowStart[g] .. rowStart[g+1] ) = sample indices of group g
//   tileStart[g] = exclusive prefix of ceil(count[g]/TM); totalTiles = sum
// ---------------------------------------------------------------------------
__global__ void setup_perm(const int* __restrict__ gid,
                           int* __restrict__ perm,
                           int* __restrict__ rowStart,
                           int* __restrict__ tileStart,
                           int* __restrict__ totalTiles) {
  __shared__ int cnt[Gn];
  __shared__ int cur[Gn];
  const int t = threadIdx.x;
  if (t < Gn) cnt[t] = 0;
  __syncthreads();
  for (int i = t; i < Bn; i += blockDim.x) atomicAdd(&cnt[gid[i]], 1);
  __syncthreads();
  if (t == 0) {
    int row = 0, tile = 0;
    for (int g = 0; g < Gn; ++g) {
      rowStart[g]  = row;
      tileStart[g] = tile;
      cur[g]       = row;
      row  += cnt[g];
      tile += (cnt[g] + TM - 1) / TM;
    }
    rowStart[Gn]  = row;
    tileStart[Gn] = tile;
    totalTiles[0] = tile;
  }
  __syncthreads();
  for (int i = t; i < Bn; i += blockDim.x) {
    const int p = atomicAdd(&cur[gid[i]], 1);
    perm[p] = i;
  }
}

// ---------------------------------------------------------------------------
// Kernel 2: fused  out = relu(Xg @ W1 + b1) @ W2 + b2  for one 16-row tile.
// routed=1: rows gathered via perm, weights indexed by this tile's group.
// routed=0: identity rows, single weight set (generic path).
// 8 waves per WG; LDS holds the gathered X tile (64KB, filled via
// GLOBAL_LOAD_ASYNC_TO_LDS_B128) and the full fp32 activation tile H (128KB)
// -- uses CDNA5's 320KB LDS/WGP.
// ---------------------------------------------------------------------------
__global__ __launch_bounds__(256, 1)
void fused_group_mlp(const float* __restrict__ X,
                     const float* __restrict__ W1, const float* __restrict__ b1,
                     const float* __restrict__ W2, const float* __restrict__ b2,
                     const int* __restrict__ perm,
                     const int* __restrict__ rowStart,
                     const int* __restrict__ tileStart,
                     const int* __restrict__ totalTiles,
                     float* __restrict__ out, int routed) {
  extern __shared__ float smem[];
  float* lX    = smem;                       // TM * LXS floats
  float* lH    = smem + TM * LXS;            // TM * LHS floats
  int*   lRows = (int*)(smem + TM * LXS + TM * LHS);  // TM ints

  const int tile = blockIdx.x;
  int g = 0, rowBase, nValid;
  if (routed) {
    if (tile >= totalTiles[0]) return;       // uniform early-exit (EXEC stays full)
    while (g + 1 < Gn && tile >= tileStart[g + 1]) ++g;
    rowBase = rowStart[g] + (tile - tileStart[g]) * TM;
    nValid  = rowStart[g + 1] - rowBase;
    if (nValid > TM) nValid = TM;
  } else {
    rowBase = tile * TM;
    nValid  = TM;
  }

  const float* W1g = W1;
  const float* b1g = b1;
  const float* W2g = W2;
  const float* b2g = b2;
  if (routed) {
    W1g += (size_t)g * IND * HID;
    b1g += (size_t)g * HID;
    W2g += (size_t)g * HID * OUTD;
    b2g += (size_t)g * OUTD;
  }

  const int t = threadIdx.x;
  if (t < TM) {
    if (routed) {
      const int r = (t < nValid) ? t : (nValid - 1);   // clamp padded rows (stores masked later)
      lRows[t] = perm[rowBase + r];
    } else {
      lRows[t] = rowBase + t;
    }
  }
  __syncthreads();

  // Gather X tile (16 x 1024 f32) straight into LDS via async b128 copies.
  // Each thread moves 16 aligned 16B chunks; ASYNCcnt tracks completion.
  for (int e = t; e < TM * (IND / 4); e += 256) {
    const int r  = e >> 8;        // IND/4 == 256 float4 per row
    const int c4 = e & 255;
    const float*   src    = X + (size_t)lRows[r] * IND + c4 * 4;
    const unsigned ldsOff = (unsigned)(uintptr_t)(lX + r * LXS + c4 * 4);
    async_load_lds_b128(ldsOff, src);
  }
  // Warm L2/L0 for the stage-2 weights while stage 1 will run (head of W2 slice).
  __builtin_prefetch(W2g + (size_t)t * 32, 0, 1);
  wait_asynccnt0();               // own wave's async copies done
  __syncthreads();                // all waves' copies visible

  const int lane = t & 31;
  const int w    = t >> 5;        // wave id 0..7
  const int half = lane >> 4;     // K-pair selector for A/B frags, M-half for C
  const int mn   = lane & 15;     // A row / B,C column within tile

  // ---------------- Stage 1: H = relu(X @ W1 + b1), H is 16 x 2048 ----------
  // Wave w owns columns [w*256, w*256+256), processed as 2 batches of 8 tiles.
  for (int nb = 0; nb < 2; ++nb) {
    const int n0 = w * 256 + nb * 128;
    v8f acc[8];
#pragma unroll
    for (int i = 0; i < 8; ++i) acc[i] = {};
    for (int k = 0; k < IND; k += 4) {
      v2f a;                                        // A: 16x4 f32 frag from LDS
      const float* ap = lX + mn * LXS + k + 2 * half;
      a.x = ap[0];
      a.y = ap[1];
#pragma unroll
      for (int ti = 0; ti < 8; ++ti) {
        const int n = n0 + ti * 16 + mn;
        v2f b;                                      // B: 4x16 f32 frag from L2-resident W1
        b.x = W1g[(size_t)(k + 2 * half)     * HID + n];
        b.y = W1g[(size_t)(k + 2 * half + 1) * HID + n];
        acc[ti] = wmma_f32(a, b, acc[ti]);
      }
    }
#pragma unroll
    for (int ti = 0; ti < 8; ++ti) {
      const int n = n0 + ti * 16 + mn;
      const float bias = b1g[n];
#pragma unroll
      for (int r = 0; r < 8; ++r) {
        const int row = r + 8 * half;
        const float v = acc[ti][r] + bias;
        lH[row * LHS + n] = v > 0.f ? v : 0.f;      // ReLU into LDS
      }
    }
  }
  __syncthreads();

  // ---------------- Stage 2: out = H @ W2 + b2, out is 16 x 256 -------------
  {
    const int n0 = w * 32;                          // wave owns 2 N-tiles
    v8f acc[2];
    acc[0] = {};
    acc[1] = {};
    for (int k = 0; k < HID; k += 4) {
      v2f a;
      const float* ap = lH + mn * LHS + k + 2 * half;
      a.x = ap[0];
      a.y = ap[1];
#pragma unroll
      for (int ti = 0; ti < 2; ++ti) {
        const int n = n0 + ti * 16 + mn;
        v2f b;
        b.x = W2g[(size_t)(k + 2 * half)     * OUTD + n];
        b.y = W2g[(size_t)(k + 2 * half + 1) * OUTD + n];
        acc[ti] = wmma_f32(a, b, acc[ti]);
      }
    }
#pragma unroll
    for (int ti = 0; ti < 2; ++ti) {
      const int n = n0 + ti * 16 + mn;
      const float bias = b2g[n];
#pragma unroll
      for (int r = 0; r < 8; ++r) {
        const int row = r + 8 * half;
        if (row < nValid) {                         // mask padded rows (stores only)
          out[(size_t)lRows[row] * OUTD + n] = acc[ti][r] + bias;
        }
      }
    }
  }
}

// ---------------------------------------------------------------------------
extern "C" void kernel_launch(void* const* d_in, const int* in_sizes, int n_in,
                              void* d_out, int out_size, void* d_ws, size_t ws_size,
                              hipStream_t stream) {
  const float* x   = (const float*)d_in[0];
  const int*   gid = (const int*)  d_in[1];
  const float* W1  = (const float*)d_in[2];
  const float* b1  = (const float*)d_in[3];
  const float* W2  = (const float*)d_in[4];
  const float* b2  = (const float*)d_in[5];
  const float* Wg1 = (const float*)d_in[6];
  const float* bg1 = (const float*)d_in[7];
  const float* Wg2 = (const float*)d_in[8];
  const float* bg2 = (const float*)d_in[9];

  float* out = (float*)d_out;                   // [B, OUT] routed output
  float* gen = out + (size_t)Bn * OUTD;         // [B, OUT] generic output

  int* wsi        = (int*)d_ws;
  int* perm       = wsi;                        // Bn ints
  int* rowStart   = wsi + Bn;                   // Gn+1
  int* tileStart  = rowStart + (Gn + 1);        // Gn+1
  int* totalTiles = tileStart + (Gn + 1);       // 1

  setup_perm<<<1, 256, 0, stream>>>(gid, perm, rowStart, tileStart, totalTiles);

  const size_t smem = (size_t)(TM * LXS + TM * LHS) * sizeof(float) + TM * sizeof(int);

  // Routed path: upper bound on tile count; excess workgroups exit early.
  const int routedTilesMax = Bn / TM + Gn;
  fused_group_mlp<<<routedTilesMax, 256, smem, stream>>>(
      x, W1, b1, W2, b2, perm, rowStart, tileStart, totalTiles, out, 1);

  // Generic path: identity rows, generic weights.
  fused_group_mlp<<<Bn / TM, 256, smem, stream>>>(
      x, Wg1, bg1, Wg2, bg2, perm, rowStart, tileStart, totalTiles, gen, 0);
}